// MMDLossFunction_14465449853181
// MI455X (gfx1250) — compile-verified
//
#include <hip/hip_runtime.h>

typedef __attribute__((ext_vector_type(2))) float v2f;
typedef __attribute__((ext_vector_type(4))) float v4f;
typedef __attribute__((ext_vector_type(8))) float v8f;

#define NROWS  128
#define DDIM   12000
#define NT     8               // 128/16 tiles per side
#define KWAVES 4               // split-K waves per tile
#define KCHUNK (DDIM / KWAVES) // 3000 floats of K per wave
#define KITERS (KCHUNK / 4)    // 750 WMMA (K=4) steps per wave

// ---------------------------------------------------------------------------
// Kernel 1: row squared-norms for X (rows 0..127) and Y (rows 128..255),
// one wave32 per row; also zero the 3 pair accumulators (stream-ordered,
// graph-capture safe).
// ---------------------------------------------------------------------------
__global__ __launch_bounds__(256) void mmd_rownorms(const float* __restrict__ X,
                                                    const float* __restrict__ Y,
                                                    float* __restrict__ norms,
                                                    float* __restrict__ acc) {
    int wid  = (int)((blockIdx.x * blockDim.x + threadIdx.x) >> 5); // 0..255
    int lane = (int)(threadIdx.x & 31);
    const float* src = (wid < NROWS) ? X : Y;
    int row = wid & (NROWS - 1);
    const float* p = src + (size_t)row * DDIM;

    float s = 0.f;
    for (int k = lane * 4; k < DDIM; k += 32 * 4) {
        v4f v = *(const v4f*)(p + k);
        s += v.x * v.x + v.y * v.y + v.z * v.z + v.w * v.w;
    }
#pragma unroll
    for (int off = 16; off > 0; off >>= 1) s += __shfl_xor(s, off, 32);
    if (lane == 0) norms[wid] = s;
    if (blockIdx.x == 0 && threadIdx.x < 3) acc[threadIdx.x] = 0.f;
}

// ---------------------------------------------------------------------------
// Kernel 2: one 16x16 Gram tile per block, 4 waves split-K, fp32 WMMA.
//   blockIdx.x = pair*64 + ti*8 + tj ; pair 0:XX 1:XY 2:YY
// A-matrix 16x4 f32 lane layout (ISA 7.12.2): lanes 0-15 hold row M=lane,
// K={0,1} in v[0],v[1]; lanes 16-31 hold K={2,3}. B (4x16) mirrors with N.
// For Gram (B = rows of b), both operands load identically.
// ---------------------------------------------------------------------------
__global__ __launch_bounds__(128) void mmd_tile(const float* __restrict__ X,
                                                const float* __restrict__ Y,
                                                const float* __restrict__ norms,
                                                float* __restrict__ acc) {
    __shared__ float ctile[KWAVES * 256];
    __shared__ float red[KWAVES];

    int pair = (int)(blockIdx.x >> 6);
    int t    = (int)(blockIdx.x & 63);
    int ti = t >> 3, tj = t & 7;

    const float* A = (pair == 2) ? Y : X;
    const float* B = (pair == 0) ? X : Y;
    int aoff = (pair == 2) ? NROWS : 0;   // which norm table for rows
    int boff = (pair == 0) ? 0 : NROWS;   // which norm table for cols

    int wave = (int)(threadIdx.x >> 5);
    int lane = (int)(threadIdx.x & 31);
    int half = lane >> 4;    // 0: K={0,1}, 1: K={2,3}
    int rsub = lane & 15;    // row within tile

    size_t kstart = (size_t)wave * KCHUNK + 2 * half;
    const float* pa = A + (size_t)(ti * 16 + rsub) * DDIM + kstart;
    const float* pb = B + (size_t)(tj * 16 + rsub) * DDIM + kstart;

    // Two independent accumulator chains to cover WMMA latency.
    v8f c0 = {}; v8f c1 = {};
    for (int i = 0; i < KITERS; i += 2) {
        v2f a0 = *(const v2f*)(pa);
        v2f b0 = *(const v2f*)(pb);
        v2f a1 = *(const v2f*)(pa + 4);
        v2f b1 = *(const v2f*)(pb + 4);
        c0 = __builtin_amdgcn_wmma_f32_16x16x4_f32(false, a0, false, b0,
                                                   (short)0, c0, false, false);
        c1 = __builtin_amdgcn_wmma_f32_16x16x4_f32(false, a1, false, b1,
                                                   (short)0, c1, false, false);
        pa += 8; pb += 8;
    }

    // Stash partial C tile: element (vgpr r, lane l) = C[r + 8*(l>>4)][l&15]
#pragma unroll
    for (int r = 0; r < 8; ++r)
        ctile[wave * 256 + r * 32 + lane] = c0[r] + c1[r];
    __syncthreads();

    // Sum split-K partials, apply exp(-d2/2), reduce the 256-element tile.
    float lsum = 0.f;
    for (int e = (int)threadIdx.x; e < 256; e += 128) {
        float s = ctile[e] + ctile[256 + e] + ctile[512 + e] + ctile[768 + e];
        int r = e >> 5, l = e & 31;
        int M  = r + 8 * (l >> 4);
        int Nn = l & 15;
        float d2 = norms[aoff + ti * 16 + M] + norms[boff + tj * 16 + Nn]
                 - 2.0f * s;
        lsum += __expf(-0.5f * d2);
    }
#pragma unroll
    for (int off = 16; off > 0; off >>= 1) lsum += __shfl_xor(lsum, off, 32);
    if (lane == 0) red[wave] = lsum;
    __syncthreads();
    if (threadIdx.x == 0)
        atomicAdd(&acc[pair], red[0] + red[1] + red[2] + red[3]);
}

// ---------------------------------------------------------------------------
// Kernel 3: combine means + avg_step term.
// ---------------------------------------------------------------------------
__global__ void mmd_final(const float* __restrict__ acc,
                          const int* __restrict__ step,
                          float* __restrict__ out) {
    if (threadIdx.x == 0 && blockIdx.x == 0) {
        const float inv = 1.0f / (128.0f * 128.0f);
        out[0] = (acc[0] - 2.0f * acc[1] + acc[2]) * inv
               + (float)step[0] * 0.002f;
    }
}

extern "C" void kernel_launch(void* const* d_in, const int* in_sizes, int n_in,
                              void* d_out, int out_size, void* d_ws, size_t ws_size,
                              hipStream_t stream) {
    const float* X    = (const float*)d_in[0];
    const float* Y    = (const float*)d_in[1];
    const int*   step = (const int*)d_in[2];
    float*       out  = (float*)d_out;

    float* norms = (float*)d_ws;     // 256 floats: [0..127]=|x|^2, [128..255]=|y|^2
    float* acc   = norms + 256;      // 3 floats: per-pair exp-sum accumulators

    // 256 rows, 8 waves per 256-thread block -> 32 blocks
    mmd_rownorms<<<32, 256, 0, stream>>>(X, Y, norms, acc);
    // 3 pairs * 64 tiles, 128 threads (4 wave32) per tile
    mmd_tile<<<3 * NT * NT, 128, 0, stream>>>(X, Y, norms, acc);
    mmd_final<<<1, 32, 0, stream>>>(acc, step, out);
}